// GeoGCN_TCN_5463198400710
// MI455X (gfx1250) — compile-verified
//
#include <hip/hip_runtime.h>

typedef float v2f __attribute__((ext_vector_type(2)));
typedef float v8f __attribute__((ext_vector_type(8)));

#define Nn   10000
#define Bb   2
#define Ll   16
#define FINf 8
#define Hh   64
#define Pp   12
#define Ee   80000

__device__ __forceinline__ v8f zero8() {
    v8f v;
#pragma unroll
    for (int i = 0; i < 8; ++i) v[i] = 0.f;
    return v;
}

// ---------------------------------------------------------------------------
// Kernel 1: input projection, only timesteps l = 14, 15 are ever needed.
// hbuf[n*256 + (b*2+li)*64 + c] = relu(b_in[c] + sum_f x[b,14+li,n,f]*W_in[f,c])
// ---------------------------------------------------------------------------
__global__ void proj_kernel(const float* __restrict__ x,
                            const float* __restrict__ W_in,
                            const float* __restrict__ b_in,
                            float* __restrict__ hbuf) {
    int t  = blockIdx.x * 256 + threadIdx.x;   // exactly N*256 threads
    int n  = t >> 8;
    int r  = t & 255;          // (b*2+li)*64 + c
    int b  = r >> 7;
    int li = (r >> 6) & 1;
    int c  = r & 63;
    int l  = 14 + li;
    const float* xr = x + (size_t)((b * Ll + l) * Nn + n) * FINf;
    float acc = b_in[c];
#pragma unroll
    for (int f = 0; f < FINf; ++f) acc += xr[f] * W_in[f * Hh + c];
    hbuf[(size_t)n * 256 + r] = fmaxf(acc, 0.f);
}

// ---------------------------------------------------------------------------
// Kernel 2: edge gather/scatter.  One 256-thread block per edge, so
// row/col/vals are wave-uniform (scalar loads); 256 fp32 atomics per edge.
// agg[row[e]*256 + f] += vals[e] * hbuf[col[e]*256 + f]
// ---------------------------------------------------------------------------
__global__ void scatter_kernel(const int* __restrict__ row,
                               const int* __restrict__ col,
                               const float* __restrict__ vals,
                               const float* __restrict__ hbuf,
                               float* __restrict__ agg) {
    int e  = blockIdx.x;
    int f  = threadIdx.x;
    int rr = row[e];
    int cc = col[e];
    float v = vals[e];
    atomicAdd(&agg[(size_t)rr * 256 + f], v * hbuf[(size_t)cc * 256 + f]);
}

// ---------------------------------------------------------------------------
// Kernel 3 (WMMA): per (b,n) row m = b*N + n:
//   z[cout]  = relu(b_tcn[cout] + sum_cin W_tcn[cout,cin,0]*h2[14,cin]
//                                + W_tcn[cout,cin,1]*h2[15,cin])        (GEMM1)
//   y[p]     = b_out[p] + sum_cout z[cout]*W_out[cout,p]                (GEMM2)
// h2 = relu(agg).  GEMM1: 16x(K=128)x64 per wave via v_wmma_f32_16x16x4_f32.
// GEMM2: 16x(K=64)x16 (cols 12..15 padded zero).
// ---------------------------------------------------------------------------
__global__ void __launch_bounds__(256)
tcn_out_wmma_kernel(const float* __restrict__ agg,
                    const float* __restrict__ W_tcn,
                    const float* __restrict__ b_tcn,
                    const float* __restrict__ W_out,
                    const float* __restrict__ b_out,
                    float* __restrict__ out) {
    __shared__ float smem[8192 + 1024];
    float* Bs = smem;           // [128][64]  K-major B for GEMM1 (32 KB)
    float* Ws = smem + 8192;    // [64][16]   padded W_out (4 KB)
    float* Zs = smem;           // reused as per-wave Z stage after barrier

    int tid = threadIdx.x;

    // Stage GEMM1 B matrix: Bs[j*64+cout] = W_tcn[cout, j&63, j>>6]
    for (int i = tid; i < 8192; i += 256) {
        int j    = i >> 6;
        int cout = i & 63;
        Bs[i] = W_tcn[(cout * Hh + (j & 63)) * 3 + (j >> 6)];
    }
    // Stage padded W_out: Ws[j2*16+p]
    for (int i = tid; i < 1024; i += 256) {
        int j2 = i >> 4;
        int p  = i & 15;
        Ws[i] = (p < Pp) ? W_out[j2 * Pp + p] : 0.f;
    }
    __syncthreads();

    int wave  = tid >> 5;
    int lane  = tid & 31;
    int lrow  = lane & 15;       // M (or N/P) index within tile
    int khalf = lane >> 4;       // selects K pair {0,1} vs {2,3}

    int tile = blockIdx.x * 8 + wave;
    if (tile > 1249) tile = 1249;     // clamp: dup waves redo last tile (same output)
    int m0 = tile * 16;
    int b  = (m0 >= Nn) ? 1 : 0;      // N divisible by 16 -> tiles never straddle b
    int n0 = m0 - b * Nn;

    v8f acc[4];
#pragma unroll
    for (int ct = 0; ct < 4; ++ct) acc[ct] = zero8();

    // A row for this lane: concat [h2(l=14,0..63) | h2(l=15,0..63)] = agg row + b*128
    const float* arow = agg + (size_t)(n0 + lrow) * 256 + b * 128;

    for (int kk = 0; kk < 32; ++kk) {
        int j = kk * 4 + khalf * 2;           // K index 0..127, 8B aligned
        v2f av = *reinterpret_cast<const v2f*>(arow + j);
        v2f a;
        a.x = fmaxf(av.x, 0.f);               // h2 = relu(agg)
        a.y = fmaxf(av.y, 0.f);
#pragma unroll
        for (int ct = 0; ct < 4; ++ct) {
            v2f bv;
            bv.x = Bs[j * 64 + ct * 16 + lrow];
            bv.y = Bs[(j + 1) * 64 + ct * 16 + lrow];
            acc[ct] = __builtin_amdgcn_wmma_f32_16x16x4_f32(
                false, a, false, bv, (short)0, acc[ct], false, false);
        }
    }

    __syncthreads();   // everyone done reading Bs; safe to overwrite with Z

    // bias + relu, stash Z (16x64) into this wave's private LDS region
    float* Zw = Zs + wave * 1024;
#pragma unroll
    for (int ct = 0; ct < 4; ++ct) {
        float bt = b_tcn[ct * 16 + lrow];
#pragma unroll
        for (int i = 0; i < 8; ++i) {
            float z = fmaxf(acc[ct][i] + bt, 0.f);
            Zw[(i + khalf * 8) * 64 + ct * 16 + lrow] = z;
        }
    }
    __syncthreads();

    // GEMM2: Y(16x16) = Z(16x64) @ Ws(64x16)
    v8f accy = zero8();
    for (int kk = 0; kk < 16; ++kk) {
        int j2 = kk * 4 + khalf * 2;
        v2f a;
        a.x = Zw[lrow * 64 + j2];
        a.y = Zw[lrow * 64 + j2 + 1];
        v2f bv;
        bv.x = Ws[j2 * 16 + lrow];
        bv.y = Ws[(j2 + 1) * 16 + lrow];
        accy = __builtin_amdgcn_wmma_f32_16x16x4_f32(
            false, a, false, bv, (short)0, accy, false, false);
    }

    // y layout: out[(b*P + p)*N + n],  p = lrow (cols 12..15 discarded)
    int p = lrow;
    if (p < Pp) {
        float bo = b_out[p];
#pragma unroll
        for (int i = 0; i < 8; ++i) {
            int n = n0 + i + khalf * 8;
            out[(size_t)(b * Pp + p) * Nn + n] = accy[i] + bo;
        }
    }
}

// ---------------------------------------------------------------------------
extern "C" void kernel_launch(void* const* d_in, const int* in_sizes, int n_in,
                              void* d_out, int out_size, void* d_ws, size_t ws_size,
                              hipStream_t stream) {
    const float* x     = (const float*)d_in[0];
    const int*   row   = (const int*)  d_in[1];
    const int*   col   = (const int*)  d_in[2];
    const float* vals  = (const float*)d_in[3];
    const float* W_in  = (const float*)d_in[4];
    const float* b_in  = (const float*)d_in[5];
    const float* W_tcn = (const float*)d_in[6];
    const float* b_tcn = (const float*)d_in[7];
    const float* W_out = (const float*)d_in[8];
    const float* b_out = (const float*)d_in[9];
    float*       out   = (float*)d_out;

    float* hbuf = (float*)d_ws;                  // N*256 floats (10.24 MB)
    float* agg  = hbuf + (size_t)Nn * 256;       // N*256 floats (10.24 MB)

    // agg accumulates via atomics -> must be zeroed every call (capture-legal).
    hipMemsetAsync(agg, 0, (size_t)Nn * 256 * sizeof(float), stream);

    proj_kernel   <<<Nn, 256, 0, stream>>>(x, W_in, b_in, hbuf);
    scatter_kernel<<<Ee, 256, 0, stream>>>(row, col, vals, hbuf, agg);
    tcn_out_wmma_kernel<<<157, 256, 0, stream>>>(agg, W_tcn, b_tcn, W_out, b_out, out);
}